// TAU_14620068676210
// MI455X (gfx1250) — compile-verified
//
#include <hip/hip_runtime.h>

// GRU over T=4096 sequence, B=512, hidden=4.
// gi = x·w_ih^T + b_ih via V_WMMA_F32_16X16X4_F32 (K=4 matches exactly).
// Recurrence: lane=(q,c), 8 batches/wave, h replicated per lane, quad
// ds_swizzle broadcast of h_new. 64 single-wave workgroups (all the
// recurrence parallelism that exists: B*C = 2048 lanes).
// Double-buffered A tiles hide global latency behind the 16-step recurrence;
// no __syncthreads (single wave: LDS ops are in-order, avoid storecnt drains).

typedef float v2f __attribute__((ext_vector_type(2)));
typedef float v8f __attribute__((ext_vector_type(8)));

#define T_SEQ 4096
#define CHUNK 16
#define NCHUNK (T_SEQ / CHUNK)
#define BPW 8             // batches per wave
#define ROW 16            // gi row stride (12 gates padded to 16, unconditional stores)
#define GI_STRIDE 260     // 16 steps * 16 padded to 260 (mod 64 = 4) for bank spread
#define OUT_STRIDE 17     // 16 + 1 pad

__device__ __forceinline__ float fast_sigmoid(float x) {
    return __builtin_amdgcn_rcpf(1.0f + __expf(-x));   // v_exp + v_rcp
}
__device__ __forceinline__ float fast_tanh(float x) {
    // tanh(x) = 1 - 2/(e^{2x}+1)
    float e = __expf(2.0f * x);
    return fmaf(-2.0f, __builtin_amdgcn_rcpf(e + 1.0f), 1.0f);
}

__global__ __launch_bounds__(32)
void gru_fused(const float* __restrict__ x,
               const float* __restrict__ w_ih,
               const float* __restrict__ w_hh,
               const float* __restrict__ b_ih,
               const float* __restrict__ b_hh,
               float* __restrict__ out) {
    __shared__ float lds_gi[BPW * GI_STRIDE];
    __shared__ float lds_out[32 * OUT_STRIDE];

    const int lane = threadIdx.x;   // 0..31
    const int n    = lane & 15;     // WMMA N column (gate) / M row within half
    const int half = lane >> 4;     // 0 or 1
    const int b0   = blockIdx.x * BPW;

    // ---- B matrix (4x16, K=channel, N=gate) for WMMA, loaded once ----
    // VGPR0 = {K=0 | K=2}, VGPR1 = {K=1 | K=3} across lane halves.
    v2f Bm;
    Bm.x = (n < 12) ? w_ih[n * 4 + 2 * half + 0] : 0.0f;
    Bm.y = (n < 12) ? w_ih[n * 4 + 2 * half + 1] : 0.0f;
    const float biasc = (n < 12) ? b_ih[n] : 0.0f;   // folded into C

    // ---- per-lane recurrence weights: lane channel c uses gate rows c, 4+c, 8+c ----
    const int c = lane & 3;
    const int q = lane >> 2;        // batch slot within wave
    const float wr0 = w_hh[c * 4 + 0], wr1 = w_hh[c * 4 + 1],
                wr2 = w_hh[c * 4 + 2], wr3 = w_hh[c * 4 + 3];
    const float wz0 = w_hh[(4 + c) * 4 + 0], wz1 = w_hh[(4 + c) * 4 + 1],
                wz2 = w_hh[(4 + c) * 4 + 2], wz3 = w_hh[(4 + c) * 4 + 3];
    const float wn0 = w_hh[(8 + c) * 4 + 0], wn1 = w_hh[(8 + c) * 4 + 1],
                wn2 = w_hh[(8 + c) * 4 + 2], wn3 = w_hh[(8 + c) * 4 + 3];
    const float bhr = b_hh[c], bhz = b_hh[4 + c], bhn = b_hh[8 + c];

    // replicated hidden state (h0..h3) + this lane's own channel value
    float h0 = 0.0f, h1 = 0.0f, h2 = 0.0f, h3 = 0.0f, hmine = 0.0f;

    // A-tile double buffer: 8 batches x (2 channels per lane half)
    v2f Abuf[BPW];
    auto load_tiles = [&](int t0) {
        #pragma unroll
        for (int j = 0; j < BPW; ++j) {
            // A 16x4: M=time (lane%16), lanes 0-15 carry K=0,1; lanes 16-31 K=2,3
            const float* xp = x + (((size_t)(b0 + j) * 4 + 2 * half) * T_SEQ) + t0 + n;
            Abuf[j].x = __builtin_nontemporal_load(xp);
            Abuf[j].y = __builtin_nontemporal_load(xp + T_SEQ);
        }
    };
    load_tiles(0);

    for (int tc = 0; tc < NCHUNK; ++tc) {
        const int t0 = tc * CHUNK;

        // ---- Phase A: gi for 8 batches x 16 steps via WMMA f32 16x16x4 ----
        #pragma unroll
        for (int j = 0; j < BPW; ++j) {
            v8f acc;
            #pragma unroll
            for (int r = 0; r < 8; ++r) acc[r] = biasc;
            acc = __builtin_amdgcn_wmma_f32_16x16x4_f32(
                /*neg_a=*/false, Abuf[j], /*neg_b=*/false, Bm,
                /*c_mod=*/(short)0, acc, /*reuse_a=*/false, /*reuse_b=*/false);
            // C layout: lane half 0 -> rows r, half 1 -> rows r+8; N = n (gate).
            // Padded 16-wide rows: store all columns unconditionally (no exec mask).
            #pragma unroll
            for (int r = 0; r < 8; ++r)
                lds_gi[j * GI_STRIDE + (r + 8 * half) * ROW + n] = acc[r];
        }

        // ---- Prefetch next chunk's A tiles; latency hidden by recurrence ----
        if (tc + 1 < NCHUNK) load_tiles(t0 + CHUNK);

        // ---- Phase B: 16 sequential GRU steps (LDS in-order within wave) ----
        #pragma unroll
        for (int tt = 0; tt < CHUNK; ++tt) {
            const float* g = &lds_gi[q * GI_STRIDE + tt * ROW];
            const float ir  = g[c];
            const float iz  = g[4 + c];
            const float inn = g[8 + c];

            const float ghr = fmaf(h3, wr3, fmaf(h2, wr2, fmaf(h1, wr1, fmaf(h0, wr0, bhr))));
            const float ghz = fmaf(h3, wz3, fmaf(h2, wz2, fmaf(h1, wz1, fmaf(h0, wz0, bhz))));
            const float ghn = fmaf(h3, wn3, fmaf(h2, wn2, fmaf(h1, wn1, fmaf(h0, wn0, bhn))));

            const float r  = fast_sigmoid(ir + ghr);
            const float z  = fast_sigmoid(iz + ghz);
            const float nn = fast_tanh(fmaf(r, ghn, inn));
            const float hnew = fmaf(z, hmine - nn, nn);   // (1-z)*n + z*h

            lds_out[(q * 4 + c) * OUT_STRIDE + tt] = hnew;

            // quad broadcast h_new from lanes (quad base + k), k=0..3
            const int u = __float_as_int(hnew);
            h0 = __int_as_float(__builtin_amdgcn_ds_swizzle(u, 0x8000 | (0 * 0x55)));
            h1 = __int_as_float(__builtin_amdgcn_ds_swizzle(u, 0x8000 | (1 * 0x55)));
            h2 = __int_as_float(__builtin_amdgcn_ds_swizzle(u, 0x8000 | (2 * 0x55)));
            h3 = __int_as_float(__builtin_amdgcn_ds_swizzle(u, 0x8000 | (3 * 0x55)));
            hmine = hnew;
        }

        // ---- Phase C: coalesced writeback (64B contiguous per half-wave) ----
        #pragma unroll
        for (int pp = 0; pp < 16; ++pp) {
            const int pair = pp * 2 + half;          // 0..31 == q2*4 + c2
            const float val = lds_out[pair * OUT_STRIDE + n];
            const int q2 = pair >> 2, c2 = pair & 3;
            __builtin_nontemporal_store(
                val, &out[((size_t)(b0 + q2) * 4 + c2) * T_SEQ + t0 + n]);
        }
        // No barrier: single wave per workgroup; LDS ops are in-order per wave,
        // so next chunk's gi stores cannot pass this chunk's reads.
    }
}

extern "C" void kernel_launch(void* const* d_in, const int* in_sizes, int n_in,
                              void* d_out, int out_size, void* d_ws, size_t ws_size,
                              hipStream_t stream) {
    (void)in_sizes; (void)n_in; (void)d_ws; (void)ws_size; (void)out_size;
    const float* x    = (const float*)d_in[0];
    const float* w_ih = (const float*)d_in[1];
    const float* w_hh = (const float*)d_in[2];
    const float* b_ih = (const float*)d_in[3];
    const float* b_hh = (const float*)d_in[4];
    float* out = (float*)d_out;
    // 512 batches / 8 per wave = 64 workgroups of one wave32 each
    gru_fused<<<64, 32, 0, stream>>>(x, w_ih, w_hh, b_ih, b_hh, out);
}